// GNNPotentials_50903952392738
// MI455X (gfx1250) — compile-verified
//
#include <hip/hip_runtime.h>
#include <hip/hip_bf16.h>
#include <math.h>

typedef __attribute__((ext_vector_type(2))) float v2f;
typedef __attribute__((ext_vector_type(8))) float v8f;

#define N_ATOMS 1536
#define FEAT    32
#define N_GAUSS 16
#define N_TYPES 10
#define CUTOFF  5.0f
#define CELL    30.0f
#define GAMMA   10.0f
#define TILE_M  16
#define NTILES  (N_ATOMS / TILE_M)   // 96
#define MU_STEP (CUTOFF / (float)(N_GAUSS - 1))
#define MAX_SLICES 16

__device__ __forceinline__ float min_image(float d) {
    // matches reference: shift = (d < -CELL/2) - (d >= CELL/2); d += shift*CELL
    float s = (d < -0.5f * CELL ? 1.0f : 0.0f) - (d >= 0.5f * CELL ? 1.0f : 0.0f);
    return d + s * CELL;
}

// Branchless tanh: tanh(x) = sign(x) * (1 - e^{-2|x|}) / (1 + e^{-2|x|}).
// Argument of exp is always <= 0 -> no overflow, no divergent slow path.
// Uses hardware v_exp_f32 + v_rcp_f32 (TRANS ops, co-execute with VALU).
__device__ __forceinline__ float fast_tanhf(float x) {
    const float ax = fabsf(x);
    const float t  = __expf(-2.0f * ax);
    const float r  = (1.0f - t) * __builtin_amdgcn_rcpf(1.0f + t);
    return copysignf(r, x);
}

// One wave32 per 16-atom i-tile per j-slice.
// agg_partial[slice][i][f] = sum_{j in slice} tanh(rbf(i,j)@W1 + b1)*mask(i,j)*h[j,f]
// via V_WMMA_F32_16X16X4_F32 (M=16 i-atoms, K=16 gaussians in 4 steps, N=2x16 features).
__global__ __launch_bounds__(32)
void gnn_pair_wmma_kernel(const float* __restrict__ xyz,
                          const int*   __restrict__ z,
                          const float* __restrict__ emb,
                          const float* __restrict__ W1,
                          const float* __restrict__ b1,
                          float* __restrict__ aggp,
                          int jper)
{
    const int lane  = threadIdx.x;      // 0..31
    const int half  = lane >> 4;        // 0 | 1
    const int n     = lane & 15;        // column / per-lane row-atom index
    const int i0    = blockIdx.x * TILE_M;
    const int slice = blockIdx.y;
    const int j0    = slice * jper;
    const int jend  = min(N_ATOMS, j0 + jper);

    // atom-type embeddings are tiny (10x32) -> LDS
    __shared__ float semb[N_TYPES * FEAT];
    for (int t = lane; t < N_TYPES * FEAT; t += 32) semb[t] = emb[t];
    __syncthreads();

    // my row atom position (lane n and lane 16+n duplicate atom i0+n)
    const int   ia = i0 + n;
    const float xi = xyz[ia * 3 + 0];
    const float yi = xyz[ia * 3 + 1];
    const float zi = xyz[ia * 3 + 2];

    // j-invariant B fragments of W1 [16 x 32], per ISA 4x16 B layout:
    // lanes 0-15 hold K = 4s+{0,1}, lanes 16-31 hold K = 4s+{2,3}, col = 16t+n
    float bxf[4][2], byf[4][2];
#pragma unroll
    for (int s = 0; s < 4; ++s) {
        const int k = 4 * s + 2 * half;
#pragma unroll
        for (int t = 0; t < 2; ++t) {
            bxf[s][t] = W1[(k + 0) * FEAT + t * 16 + n];
            byf[s][t] = W1[(k + 1) * FEAT + t * 16 + n];
        }
    }
    const float b1v0 = b1[n];
    const float b1v1 = b1[16 + n];

    float acc0[8], acc1[8];
#pragma unroll
    for (int r = 0; r < 8; ++r) { acc0[r] = 0.0f; acc1[r] = 0.0f; }

    for (int j = j0; j < jend; ++j) {
        // disp[i,j] = xyz[j] - xyz[i], minimum-image wrapped
        const float dx = min_image(xyz[j * 3 + 0] - xi);
        const float dy = min_image(xyz[j * 3 + 1] - yi);
        const float dz = min_image(xyz[j * 3 + 2] - zi);
        const float d2 = dx * dx + dy * dy + dz * dz;
        const bool  msk = (d2 < CUTOFF * CUTOFF) && (d2 > 0.0f);
        const unsigned mb = (unsigned)__ballot(msk) & 0xFFFFu;  // row masks (lanes 16-31 duplicate)
        if (mb == 0u) continue;   // ~98% of j pruned; ballot-uniform branch keeps EXEC all-1 for WMMA

        const float d = sqrtf(d2);

        v8f c0 = {}; v8f c1 = {};
#pragma unroll
        for (int s = 0; s < 4; ++s) {
            const int   k0 = 4 * s + 2 * half;
            const float t0 = d - (float)(k0 + 0) * MU_STEP;
            const float t1 = d - (float)(k0 + 1) * MU_STEP;
            v2f a;
            a.x = msk ? __expf(-GAMMA * t0 * t0) : 0.0f;   // rbf * mask
            a.y = msk ? __expf(-GAMMA * t1 * t1) : 0.0f;
            v2f bt;
            bt.x = bxf[s][0]; bt.y = byf[s][0];
            c0 = __builtin_amdgcn_wmma_f32_16x16x4_f32(false, a, false, bt, (short)0, c0, false, false);
            bt.x = bxf[s][1]; bt.y = byf[s][1];
            c1 = __builtin_amdgcn_wmma_f32_16x16x4_f32(false, a, false, bt, (short)0, c1, false, false);
        }

        // h[j, f] = emb[z[j], f] from LDS; accumulate agg directly in D layout
        const int   zj  = z[j];
        const float hj0 = semb[zj * FEAT + n];
        const float hj1 = semb[zj * FEAT + 16 + n];
#pragma unroll
        for (int r = 0; r < 8; ++r) {
            const int   M  = r + 8 * half;                 // D-tile row -> i-atom i0+M
            const float rm = (float)((mb >> M) & 1u);      // mask after tanh (matches ref)
            acc0[r] += fast_tanhf(c0[r] + b1v0) * rm * hj0;
            acc1[r] += fast_tanhf(c1[r] + b1v1) * rm * hj1;
        }
    }

    float* out = aggp + (size_t)slice * N_ATOMS * FEAT;
#pragma unroll
    for (int r = 0; r < 8; ++r) {
        const int M = r + 8 * half;
        out[(i0 + M) * FEAT + n]      = acc0[r];
        out[(i0 + M) * FEAT + 16 + n] = acc1[r];
    }
}

// Per-atom readout: x = h + sum_slices(agg); u = silu(x@W2 + b2); e = u@W3 + b3.
// Deterministic in-block tree reduction -> one partial per block.
__global__ __launch_bounds__(256)
void gnn_readout_kernel(const int*   __restrict__ z,
                        const float* __restrict__ emb,
                        const float* __restrict__ W2,
                        const float* __restrict__ b2,
                        const float* __restrict__ W3,
                        const float* __restrict__ b3,
                        const float* __restrict__ aggp,
                        int nslice,
                        float* __restrict__ partial)
{
    __shared__ float red[256];
    const int i = blockIdx.x * 256 + threadIdx.x;
    float e = 0.0f;
    if (i < N_ATOMS) {
        float x[FEAT];
        const int zi = z[i];
#pragma unroll
        for (int f = 0; f < FEAT; ++f) x[f] = emb[zi * FEAT + f];
        for (int s = 0; s < nslice; ++s) {
            const float* a = aggp + ((size_t)s * N_ATOMS + i) * FEAT;
#pragma unroll
            for (int f = 0; f < FEAT; ++f) x[f] += a[f];
        }
        float ei = b3[0];
        for (int o = 0; o < FEAT; ++o) {
            float p = b2[o];
#pragma unroll
            for (int f = 0; f < FEAT; ++f) p += x[f] * W2[f * FEAT + o];
            const float u = p * __builtin_amdgcn_rcpf(1.0f + __expf(-p));   // silu
            ei += u * W3[o];
        }
        e = ei;
    }
    red[threadIdx.x] = e;
    __syncthreads();
    for (int off = 128; off > 0; off >>= 1) {
        if ((int)threadIdx.x < off) red[threadIdx.x] += red[threadIdx.x + off];
        __syncthreads();
    }
    if (threadIdx.x == 0) partial[blockIdx.x] = red[0];
}

__global__ void gnn_finalize_kernel(const float* __restrict__ partial, int nparts,
                                    float* __restrict__ out)
{
    if (threadIdx.x == 0) {
        float s = 0.0f;
        for (int p = 0; p < nparts; ++p) s += partial[p];
        out[0] = s;   // plain store, no atomics -> deterministic, no pre-zeroing needed
    }
}

extern "C" void kernel_launch(void* const* d_in, const int* in_sizes, int n_in,
                              void* d_out, int out_size, void* d_ws, size_t ws_size,
                              hipStream_t stream) {
    const float* xyz = (const float*)d_in[0];
    const int*   z   = (const int*)  d_in[1];
    const float* emb = (const float*)d_in[2];
    const float* W1  = (const float*)d_in[3];
    const float* b1  = (const float*)d_in[4];
    const float* W2  = (const float*)d_in[5];
    const float* b2  = (const float*)d_in[6];
    const float* W3  = (const float*)d_in[7];
    const float* b3  = (const float*)d_in[8];

    // workspace: nslice * [N_ATOMS x FEAT] agg partials + per-block readout partials
    const size_t slice_bytes = (size_t)N_ATOMS * FEAT * sizeof(float);
    int nslice = MAX_SLICES;
    while (nslice > 1 && (size_t)nslice * slice_bytes + 64 > ws_size) nslice >>= 1;
    float* aggp    = (float*)d_ws;
    float* partial = (float*)((char*)d_ws + (size_t)nslice * slice_bytes);
    const int jper = (N_ATOMS + nslice - 1) / nslice;

    dim3 g1(NTILES, nslice);
    gnn_pair_wmma_kernel<<<g1, 32, 0, stream>>>(xyz, z, emb, W1, b1, aggp, jper);

    const int RB = 256;
    const int nblocks = (N_ATOMS + RB - 1) / RB;  // 6
    gnn_readout_kernel<<<nblocks, RB, 0, stream>>>(z, emb, W2, b2, W3, b3, aggp, nslice, partial);
    gnn_finalize_kernel<<<1, 32, 0, stream>>>(partial, nblocks, (float*)d_out);
}